// GATAnomalyDetector_4853313045131
// MI455X (gfx1250) — compile-verified
//
#include <hip/hip_runtime.h>
#include <hip/hip_bf16.h>

#define N_NODES   50000
#define N_EDGES   1600000
#define CH        128
#define NEG_SLOPE 0.2f
#define ET        (N_EDGES + N_NODES)   // edges + self loops

typedef float v2f __attribute__((ext_vector_type(2)));
typedef float v8f __attribute__((ext_vector_type(8)));

// ---------------------------------------------------------------------------
// order-preserving float<->uint mapping for atomic float max
// ---------------------------------------------------------------------------
__device__ inline unsigned f2ord(float f) {
  unsigned u = __float_as_uint(f);
  return (u & 0x80000000u) ? ~u : (u | 0x80000000u);
}
__device__ inline float ord2f(unsigned u) {
  return (u & 0x80000000u) ? __uint_as_float(u ^ 0x80000000u)
                           : __uint_as_float(~u);
}

// ---------------------------------------------------------------------------
// One 16x16 fp32 tile of C = A(16xK) * B(Kx16) via V_WMMA_F32_16X16X4_F32.
// A row-major (lda), B row-major (ldb). Per ISA 7.12.2:
//   A 16x4 : lanes 0-15 rows M, VGPR0 = K{0|2}, VGPR1 = K{1|3} (halves by lane)
//   B 4x16 : rows striped across lanes in a VGPR, K halves by lane group
//   C/D    : VGPR i -> row (i + 8*half), col = lane&15
// ---------------------------------------------------------------------------
__device__ inline v8f gemm_tile_f32(const float* __restrict__ A, int lda,
                                    const float* __restrict__ B, int ldb,
                                    int K) {
  const int lane = threadIdx.x & 31;
  const int half = lane >> 4;   // 0: lanes 0-15, 1: lanes 16-31
  const int m    = lane & 15;
  v8f c = {0.f, 0.f, 0.f, 0.f, 0.f, 0.f, 0.f, 0.f};
#pragma unroll 4
  for (int k0 = 0; k0 < K; k0 += 4) {
    const float* ap = A + (size_t)m * lda + k0 + 2 * half;
    v2f a;
    a.x = ap[0];
    a.y = ap[1];
    const float* bp = B + (size_t)(k0 + 2 * half) * ldb + m;
    v2f b;
    b.x = bp[0];
    b.y = bp[ldb];
    c = __builtin_amdgcn_wmma_f32_16x16x4_f32(false, a, false, b, (short)0, c,
                                              false, false);
  }
  return c;
}

// ---------------------------------------------------------------------------
// Kernel 1: h = x @ W   (N_NODES x CH) = (N_NODES x CH) * (CH x CH)
// one wave per 16x16 tile; grid sized exactly (no partial waves -> EXEC all 1)
// ---------------------------------------------------------------------------
__global__ void k_gemm_h(const float* __restrict__ x,
                         const float* __restrict__ W,
                         float* __restrict__ h) {
  const int w  = (blockIdx.x * blockDim.x + threadIdx.x) >> 5;
  const int mt = w >> 3;        // CH/16 == 8 tiles in N
  const int nt = w & 7;
  const int row0 = mt * 16, n0 = nt * 16;
  v8f c = gemm_tile_f32(x + (size_t)row0 * CH, CH, W + n0, CH, CH);
  const int lane = threadIdx.x & 31;
  const int half = lane >> 4;
  const int n    = lane & 15;
#pragma unroll
  for (int i = 0; i < 8; ++i)
    h[(size_t)(row0 + i + 8 * half) * CH + n0 + n] = c[i];
}

// ---------------------------------------------------------------------------
// Kernel 2: e_src[n] = h[n,:] . a_src ; e_dst[n] = h[n,:] . a_dst
// one wave per node, 4 channels/lane, shfl_xor reduction (wave32)
// ---------------------------------------------------------------------------
__global__ void k_attn(const float* __restrict__ h,
                       const float* __restrict__ a_src,
                       const float* __restrict__ a_dst,
                       float* __restrict__ es, float* __restrict__ ed) {
  const int n    = (blockIdx.x * blockDim.x + threadIdx.x) >> 5;
  const int lane = threadIdx.x & 31;
  const float4 hv = *(const float4*)(h + (size_t)n * CH + lane * 4);
  const float4 as = *(const float4*)(a_src + lane * 4);
  const float4 ad = *(const float4*)(a_dst + lane * 4);
  float s = hv.x * as.x + hv.y * as.y + hv.z * as.z + hv.w * as.w;
  float d = hv.x * ad.x + hv.y * ad.y + hv.z * ad.z + hv.w * ad.w;
#pragma unroll
  for (int off = 16; off; off >>= 1) {
    s += __shfl_xor(s, off, 32);
    d += __shfl_xor(d, off, 32);
  }
  if (lane == 0) { es[n] = s; ed[n] = d; }
}

// ---------------------------------------------------------------------------
// Kernel 3: init accumulators (z half of d_out, denom, running-max keys)
// ---------------------------------------------------------------------------
__global__ void k_init(float* __restrict__ zacc, float* __restrict__ denom,
                       unsigned* __restrict__ mkey) {
  const int i = blockIdx.x * blockDim.x + threadIdx.x;
  if (i < N_NODES * CH) zacc[i] = 0.f;
  if (i < N_NODES) { denom[i] = 0.f; mkey[i] = 0u; }
}

// ---------------------------------------------------------------------------
// Kernel 4: per-edge leaky-relu logits + segment max over dst
// ---------------------------------------------------------------------------
__global__ void k_edge_logits(const long long* __restrict__ ei,
                              const float* __restrict__ es,
                              const float* __restrict__ ed,
                              float* __restrict__ logits,
                              unsigned* __restrict__ mkey) {
  const int e = blockIdx.x * blockDim.x + threadIdx.x;
  if (e >= ET) return;
  long long s, d;
  if (e < N_EDGES) { s = ei[e]; d = ei[N_EDGES + e]; }
  else             { s = d = (long long)(e - N_EDGES); }   // self loop
  float lg = es[s] + ed[d];
  lg = (lg > 0.f) ? lg : NEG_SLOPE * lg;
  logits[e] = lg;
  atomicMax(&mkey[d], f2ord(lg));
}

// ---------------------------------------------------------------------------
// Kernel 5: ex = exp(logit - max[dst]) (in-place) + segment sum of ex
// ---------------------------------------------------------------------------
__global__ void k_edge_exp(const long long* __restrict__ ei,
                           const unsigned* __restrict__ mkey,
                           float* __restrict__ logits,
                           float* __restrict__ denom) {
  const int e = blockIdx.x * blockDim.x + threadIdx.x;
  if (e >= ET) return;
  const long long d = (e < N_EDGES) ? ei[N_EDGES + e]
                                    : (long long)(e - N_EDGES);
  const float ex = expf(logits[e] - ord2f(mkey[d]));
  logits[e] = ex;
  atomicAdd(&denom[d], ex);
}

// ---------------------------------------------------------------------------
// Kernel 6: z[dst,:] += (ex/denom[dst]) * h[src,:]
// one wave per edge, 4 ch/lane; h and z are L2-resident (25.6 MB each << 192MB)
// ---------------------------------------------------------------------------
__global__ void k_edge_scatter(const long long* __restrict__ ei,
                               const float* __restrict__ h,
                               const float* __restrict__ logits,
                               const float* __restrict__ denom,
                               float* __restrict__ zacc) {
  const int e = (blockIdx.x * blockDim.x + threadIdx.x) >> 5;
  if (e >= ET) return;
  const int lane = threadIdx.x & 31;
  long long s, d;
  if (e < N_EDGES) { s = ei[e]; d = ei[N_EDGES + e]; }
  else             { s = d = (long long)(e - N_EDGES); }
  const float coef = logits[e] / denom[d];
  const float4 hv = *(const float4*)(h + (size_t)s * CH + lane * 4);
  float* zp = zacc + (size_t)d * CH + lane * 4;
  atomicAdd(zp + 0, coef * hv.x);
  atomicAdd(zp + 1, coef * hv.y);
  atomicAdd(zp + 2, coef * hv.z);
  atomicAdd(zp + 3, coef * hv.w);
}

// ---------------------------------------------------------------------------
// Kernel 7: z += b_gat (in place, z lives in second half of d_out)
// ---------------------------------------------------------------------------
__global__ void k_bias(float* __restrict__ z, const float* __restrict__ b_gat) {
  const int i = blockIdx.x * blockDim.x + threadIdx.x;
  if (i < N_NODES * CH) z[i] += b_gat[i & (CH - 1)];
}

// ---------------------------------------------------------------------------
// Kernel 8: recon = tanh(z @ Wd + bd), WMMA tiles as kernel 1
// ---------------------------------------------------------------------------
__global__ void k_gemm_recon(const float* __restrict__ z,
                             const float* __restrict__ Wd,
                             const float* __restrict__ bd,
                             float* __restrict__ recon) {
  const int w  = (blockIdx.x * blockDim.x + threadIdx.x) >> 5;
  const int mt = w >> 3;
  const int nt = w & 7;
  const int row0 = mt * 16, n0 = nt * 16;
  v8f c = gemm_tile_f32(z + (size_t)row0 * CH, CH, Wd + n0, CH, CH);
  const int lane = threadIdx.x & 31;
  const int half = lane >> 4;
  const int n    = lane & 15;
  const float bias = bd[n0 + n];
#pragma unroll
  for (int i = 0; i < 8; ++i)
    recon[(size_t)(row0 + i + 8 * half) * CH + n0 + n] = tanhf(c[i] + bias);
}

// ---------------------------------------------------------------------------
extern "C" void kernel_launch(void* const* d_in, const int* in_sizes, int n_in,
                              void* d_out, int out_size, void* d_ws,
                              size_t ws_size, hipStream_t stream) {
  const float*     x     = (const float*)d_in[0];
  const long long* ei    = (const long long*)d_in[1];  // int64 [2, E]
  const float*     W     = (const float*)d_in[2];
  const float*     a_src = (const float*)d_in[3];
  const float*     a_dst = (const float*)d_in[4];
  const float*     b_gat = (const float*)d_in[5];
  const float*     Wd    = (const float*)d_in[6];
  const float*     bd    = (const float*)d_in[7];

  float* out   = (float*)d_out;                    // [recon | z]
  float* recon = out;
  float* z     = out + (size_t)N_NODES * CH;       // z accumulated in place

  // workspace layout (~33 MB)
  float*    h      = (float*)d_ws;                         // N*CH
  float*    es     = h + (size_t)N_NODES * CH;             // N
  float*    ed     = es + N_NODES;                         // N
  float*    denom  = ed + N_NODES;                         // N
  unsigned* mkey   = (unsigned*)(denom + N_NODES);         // N
  float*    logits = (float*)(mkey + N_NODES);             // ET

  const dim3 blk(256);
  const int tilesWaves = (N_NODES / 16) * (CH / 16);       // 25000 waves
  k_gemm_h     <<<tilesWaves / 8,              blk, 0, stream>>>(x, W, h);
  k_attn       <<<N_NODES / 8,                 blk, 0, stream>>>(h, a_src, a_dst, es, ed);
  k_init       <<<(N_NODES * CH + 255) / 256,  blk, 0, stream>>>(z, denom, mkey);
  k_edge_logits<<<(ET + 255) / 256,            blk, 0, stream>>>(ei, es, ed, logits, mkey);
  k_edge_exp   <<<(ET + 255) / 256,            blk, 0, stream>>>(ei, mkey, logits, denom);
  k_edge_scatter<<<ET / 8,                     blk, 0, stream>>>(ei, h, logits, denom, z);
  k_bias       <<<(N_NODES * CH + 255) / 256,  blk, 0, stream>>>(z, b_gat);
  k_gemm_recon <<<tilesWaves / 8,              blk, 0, stream>>>(z, Wd, bd, recon);
}